// DiffAugment_78477642432692
// MI455X (gfx1250) — compile-verified
//
#include <hip/hip_runtime.h>
#include <hip/hip_bf16.h>
#include <stdint.h>

// ---------------------------------------------------------------------------
// DiffAugment fused kernels for MI455X (gfx1250).
// Shapes fixed by reference: B=128, C=3, H=W=256, shift=32, cut=51.
// Bandwidth-bound: ~200MB HBM traffic -> ~8.6us floor @ 23.3TB/s.
// CDNA5 paths: GLOBAL_LOAD_ASYNC_TO_LDS_B128 + s_wait_asynccnt (row staging),
// global_prefetch_b8 (streaming reduction), NT store hints (output stream so
// the L2-resident input survives pass 2).
// ---------------------------------------------------------------------------

#define B_    128
#define C_    3
#define H_    256
#define W_    256
#define HW_   (H_ * W_)
#define SHIFT_ 32
#define HALF_  25   // cut=51 -> cut//2

typedef int v4i_ __attribute__((ext_vector_type(4)));
#define AS1_ __attribute__((address_space(1)))
#define AS3_ __attribute__((address_space(3)))

#if defined(__has_builtin)
#  if __has_builtin(__builtin_amdgcn_global_load_async_to_lds_b128)
#    define HAVE_ASYNC_LDS 1
#  endif
#endif
#ifndef HAVE_ASYNC_LDS
#  define HAVE_ASYNC_LDS 0
#  warning "gfx1250 async-to-LDS builtin not found; using direct vector loads"
#endif

#if defined(__has_builtin)
#  if __has_builtin(__builtin_amdgcn_s_wait_asynccnt)
#    define WAIT_ASYNC0() __builtin_amdgcn_s_wait_asynccnt(0)
#  endif
#endif
#ifndef WAIT_ASYNC0
#  define WAIT_ASYNC0() asm volatile("s_wait_asynccnt 0x0" ::: "memory")
#endif

// ---------------------------------------------------------------------------
// Pass 1: per-batch mean over (C,H,W).  128 blocks x 256 threads.
// Streams the whole input once with regular-temporal policy so it lands (and
// stays) in the 192MB L2 for pass 2's re-read.  Deterministic LDS tree
// reduction (no float atomics -> bitwise-stable across replays).
// ---------------------------------------------------------------------------
__global__ __launch_bounds__(256) void batch_mean_kernel(
    const float* __restrict__ x, float* __restrict__ M) {
  const int b = blockIdx.x;
  const int t = threadIdx.x;
  const int NF4 = (C_ * HW_) / 4;          // 49152 float4 per batch
  const float4* __restrict__ p =
      reinterpret_cast<const float4*>(x) + (size_t)b * NF4;

  float s = 0.0f;
#pragma unroll 8
  for (int i = 0; i < NF4 / 256; ++i) {    // 192 iterations
    // Locality 3 -> near-scope regular-temporal prefetch: pulls lines into
    // all cache levels (ISA 10.5) so they persist in L2 for pass 2.
    // Speculative prefetch past the allocation is silently dropped.
    __builtin_prefetch(p + (size_t)(i + 8) * 256 + t, 0, 3);
    float4 v = p[(size_t)i * 256 + t];
    s += (v.x + v.y) + (v.z + v.w);
  }

  __shared__ float red[256];
  red[t] = s;
  __syncthreads();
#pragma unroll
  for (int off = 128; off > 0; off >>= 1) {
    if (t < off) red[t] += red[t + off];
    __syncthreads();
  }
  if (t == 0) M[b] = red[0] * (1.0f / (float)(C_ * HW_));
}

// ---------------------------------------------------------------------------
// Pass 2: fused color-transform + translation gather + cutout.
// One block per (b, h) output row; 256 threads = one thread per output w.
// The (block-uniform) source row for all 3 channels is staged into LDS via
// async b128 copies (threads 0..191 move 16B each, ASYNCcnt-tracked), then
// each lane computes out_c = A*x_c + E*mc + D for its pixel.  Output rows are
// written with non-temporal stores: never re-read, keep input in L2.
// ---------------------------------------------------------------------------
__global__ __launch_bounds__(256) void augment_kernel(
    const float* __restrict__ x,
    const float* __restrict__ r_bright,
    const float* __restrict__ r_sat,
    const float* __restrict__ r_con,
    const int*   __restrict__ t_x,
    const int*   __restrict__ t_y,
    const int*   __restrict__ off_x,
    const int*   __restrict__ off_y,
    const float* __restrict__ M,
    float*       __restrict__ out) {
  const int bh = blockIdx.x;
  const int b  = bh >> 8;          // / H_
  const int h  = bh & (H_ - 1);
  const int w  = threadIdx.x;

  // Per-batch scalars (block-uniform -> scalar loads).
  const float k = r_con[b] + 0.5f;            // contrast gain
  const float A = k * (2.0f * r_sat[b]);      // net per-element gain
  const float E = k - A;                      // channel-mean coefficient
  const float D = M[b] * (1.0f - k) + (r_bright[b] - 0.5f);
  const int tx = t_x[b] - SHIFT_;
  const int ty = t_y[b] - SHIFT_;
  const int ox = off_x[b], oy = off_y[b];
  const int xlo = max(0, ox - HALF_), xhi = min(H_ - 1, ox + HALF_);
  const int ylo = max(0, oy - HALF_), yhi = min(W_ - 1, oy + HALF_);

  const int  sh   = h + tx;                        // source row
  const bool vrow = ((unsigned)sh < (unsigned)H_); // uniform per block

  __shared__ float row[C_ * W_];   // 3KB: three source channel rows

  if (vrow && w < 192) {
    const int c = w >> 6;          // channel 0..2
    const int q = w & 63;          // 16-byte chunk within the row
    const float* gsrc = x + ((size_t)(b * C_ + c)) * HW_
                          + (size_t)sh * W_ + (size_t)q * 4;
    float* ldst = &row[c * W_ + q * 4];
#if HAVE_ASYNC_LDS
    // Param 0: v4i in AS1 (global src); param 1: v4i in AS3 (LDS dst);
    // then imm offset, imm cpol (0 = RT so pass-1-resident L2 lines hit).
    __builtin_amdgcn_global_load_async_to_lds_b128(
        (AS1_ v4i_*)(v4i_*)gsrc,
        (AS3_ v4i_*)(v4i_*)ldst,
        0, 0);
#else
    *(float4*)ldst = *(const float4*)gsrc;
#endif
  }
#if HAVE_ASYNC_LDS
  WAIT_ASYNC0();                   // own wave's async copies landed in LDS
#endif
  __syncthreads();                 // make all waves' rows visible

  float o0 = 0.0f, o1 = 0.0f, o2 = 0.0f;
  const int  sw  = w + ty;
  const bool cut = (h >= xlo) & (h <= xhi) & (w >= ylo) & (w <= yhi);
  if (vrow && ((unsigned)sw < (unsigned)W_) && !cut) {
    const float x0 = row[sw];
    const float x1 = row[W_ + sw];
    const float x2 = row[2 * W_ + sw];
    const float mc = (x0 + x1 + x2) * (1.0f / 3.0f);
    o0 = A * x0 + E * mc + D;
    o1 = A * x1 + E * mc + D;
    o2 = A * x2 + E * mc + D;
  }
  const size_t ob = (size_t)(b * C_) * HW_ + (size_t)h * W_ + (size_t)w;
  // Non-temporal: output is write-once, don't evict the L2-resident input.
  __builtin_nontemporal_store(o0, out + ob);
  __builtin_nontemporal_store(o1, out + ob + HW_);
  __builtin_nontemporal_store(o2, out + ob + 2 * HW_);
}

// ---------------------------------------------------------------------------
extern "C" void kernel_launch(void* const* d_in, const int* in_sizes, int n_in,
                              void* d_out, int out_size, void* d_ws, size_t ws_size,
                              hipStream_t stream) {
  const float* x  = (const float*)d_in[0];
  const float* rb = (const float*)d_in[1];
  const float* rs = (const float*)d_in[2];
  const float* rc = (const float*)d_in[3];
  const int*   tx = (const int*)d_in[4];
  const int*   ty = (const int*)d_in[5];
  const int*   ox = (const int*)d_in[6];
  const int*   oy = (const int*)d_in[7];
  float* out = (float*)d_out;
  float* M   = (float*)d_ws;      // 128 floats of scratch

  batch_mean_kernel<<<B_, 256, 0, stream>>>(x, M);
  augment_kernel<<<B_ * H_, 256, 0, stream>>>(x, rb, rs, rc, tx, ty, ox, oy, M, out);
}